// multiclassdice_88115549045609
// MI455X (gfx1250) — compile-verified
//
#include <hip/hip_runtime.h>

#define SMOOTH 1e-5f

typedef __attribute__((ext_vector_type(2))) float v2f;
typedef __attribute__((ext_vector_type(8))) float v8f;

// ---------------------------------------------------------------------------
// Kernel 0: zero the 16-float accumulator workspace (sum_x[5], inter[5], cnt[5], pad)
// ---------------------------------------------------------------------------
__global__ __launch_bounds__(32) void dice_init_kernel(float* __restrict__ ws) {
    if (threadIdx.x < 16) ws[threadIdx.x] = 0.0f;
}

// ---------------------------------------------------------------------------
// Kernel 1: streaming reduction.
//   input  x : [4,10,5,384,384] f32  (per (n,t): 5 channel planes of 147456)
//   target t : [4,10,1,384,384] f32
// Each thread accumulates sum_x[5], inter[5], cnt[5] over a grid-stride loop of
// float4 units; intra-wave reduce via shfl_xor; cross-wave reduce via
// V_WMMA_F32_16X16X4_F32 (A = ones -> column sums); 15 atomics per block.
// ---------------------------------------------------------------------------
__global__ __launch_bounds__(256) void dice_main_kernel(const float* __restrict__ x,
                                                        const float* __restrict__ tgt,
                                                        float* __restrict__ ws) {
    constexpr int IMG4 = (384 * 384) / 4;   // 36864 float4 per channel plane
    constexpr int PV   = 40 * IMG4;         // 1,474,560 float4 target units

    float sumx[5]  = {0.f, 0.f, 0.f, 0.f, 0.f};
    float inter[5] = {0.f, 0.f, 0.f, 0.f, 0.f};
    float cnt[5]   = {0.f, 0.f, 0.f, 0.f, 0.f};

    const float4* __restrict__ t4 = (const float4*)tgt;
    const float4* __restrict__ x4 = (const float4*)x;

    const int stride = (int)(gridDim.x * blockDim.x);
    for (int q = (int)(blockIdx.x * blockDim.x + threadIdx.x); q < PV; q += stride) {
        const int nt = q / IMG4;               // which (n,t) image
        const int r  = q - nt * IMG4;          // float4 offset within plane
        const int xb = nt * 5 * IMG4 + r;      // class-0 plane, same pixel

        float4 tv = t4[q];
        float4 xv[5];
#pragma unroll
        for (int c = 0; c < 5; ++c) xv[c] = x4[xb + c * IMG4];

        // Speculative prefetch of next grid-stride iteration (global_prefetch_b8).
        if (q + stride < PV) {
            __builtin_prefetch(&t4[q + stride], 0, 1);
            __builtin_prefetch(&x4[xb + stride], 0, 1);
        }

        const float ta[4] = {tv.x, tv.y, tv.z, tv.w};
#pragma unroll
        for (int e = 0; e < 4; ++e) {
            const float t = ta[e];
            const int cls = (t >= 0.25f) + (t >= 0.375f) + (t >= 0.5f) + (t >= 0.625f);
#pragma unroll
            for (int c = 0; c < 5; ++c) {
                const float4 xc4 = xv[c];
                const float xa[4] = {xc4.x, xc4.y, xc4.z, xc4.w};
                const float xe = xa[e];
                const float m  = (cls == c) ? 1.0f : 0.0f;
                sumx[c]  += xe;
                inter[c]  = fmaf(m, xe, inter[c]);
                cnt[c]   += m;
            }
        }
    }

    // ---- intra-wave reduction (wave32) ----
#pragma unroll
    for (int i = 0; i < 5; ++i) {
#pragma unroll
        for (int off = 16; off > 0; off >>= 1) {
            sumx[i]  += __shfl_xor(sumx[i],  off, 32);
            inter[i] += __shfl_xor(inter[i], off, 32);
            cnt[i]   += __shfl_xor(cnt[i],   off, 32);
        }
    }

    // ---- cross-wave reduction via WMMA ----
    // part[w][0..4]=sum_x, [5..9]=inter, [10..14]=cnt, [15]=0 pad; 8 waves/block.
    __shared__ float part[8][16];
    const int wave = (int)(threadIdx.x >> 5);
    const int lane = (int)(threadIdx.x & 31);
    if (lane == 0) {
#pragma unroll
        for (int i = 0; i < 5; ++i) {
            part[wave][i]      = sumx[i];
            part[wave][5 + i]  = inter[i];
            part[wave][10 + i] = cnt[i];
        }
        part[wave][15] = 0.0f;
    }
    __syncthreads();

    if (threadIdx.x < 32) {   // wave 0, EXEC all-ones (WMMA requirement)
        const int col = lane & 15;
        const int hi  = lane >> 4;
        v2f a;  a[0] = 1.0f;  a[1] = 1.0f;          // A = ones(16x4) -> D = colsum(B)
        v8f acc = {0.f, 0.f, 0.f, 0.f, 0.f, 0.f, 0.f, 0.f};
#pragma unroll
        for (int t = 0; t < 2; ++t) {
            // Column `col` of B gets the 4 wave-partials {4t,4t+1,4t+2,4t+3}[col],
            // distributed over the 4 k-slots (order irrelevant for a sum).
            v2f b;
            b[0] = part[4 * t + hi][col];
            b[1] = part[4 * t + 2 + hi][col];
            acc = __builtin_amdgcn_wmma_f32_16x16x4_f32(
                /*neg_a=*/false, a, /*neg_b=*/false, b,
                /*c_mod=*/(short)0, acc, /*reuse_a=*/false, /*reuse_b=*/false);
        }
        // Row m=0 of D lives in acc[0] of lanes 0..15: the 16 column totals.
        if (lane < 15) atomicAdd(&ws[lane], acc[0]);
    }
}

// ---------------------------------------------------------------------------
// Kernel 2: epilogue — weighted dice loss from the 15 global sums.
// ---------------------------------------------------------------------------
__global__ __launch_bounds__(32) void dice_final_kernel(const float* __restrict__ ws,
                                                        const float* __restrict__ wgt,
                                                        float* __restrict__ out) {
    if (threadIdx.x == 0) {
        float loss = 0.0f;
#pragma unroll
        for (int c = 0; c < 5; ++c) {
            const float sx = ws[c];
            const float it = ws[5 + c];
            const float cn = ws[10 + c];
            const float dice = 1.0f - (2.0f * it + SMOOTH) / (sx + cn + SMOOTH);
            loss += wgt[c] * dice;
        }
        out[0] = loss;
    }
}

// ---------------------------------------------------------------------------
extern "C" void kernel_launch(void* const* d_in, const int* in_sizes, int n_in,
                              void* d_out, int out_size, void* d_ws, size_t ws_size,
                              hipStream_t stream) {
    (void)in_sizes; (void)n_in; (void)out_size; (void)ws_size;
    const float* x   = (const float*)d_in[0];   // [4,10,5,384,384]
    const float* tgt = (const float*)d_in[1];   // [4,10,1,384,384]
    const float* wgt = (const float*)d_in[2];   // [5]
    float*       ws  = (float*)d_ws;            // 16 floats of scratch
    float*       out = (float*)d_out;           // scalar f32

    dice_init_kernel<<<1, 32, 0, stream>>>(ws);
    dice_main_kernel<<<2048, 256, 0, stream>>>(x, tgt, ws);
    dice_final_kernel<<<1, 32, 0, stream>>>(ws, wgt, out);
}